// ParagraphVector_20478404067832
// MI455X (gfx1250) — compile-verified
//
#include <hip/hip_runtime.h>
#include <hip/hip_bf16.h>
#include <math.h>

// ParagraphVector forward for MI455X (gfx1250, wave32).
// Per b: gather review emb, 60 dot products vs. it via V_WMMA_F32_16X16X4_F32
// (f32 in/out -> reference-precision), softplus + masked mean.

typedef __attribute__((ext_vector_type(2))) float v2f;
typedef __attribute__((ext_vector_type(8))) float v8f;

#define EMB   128
#define WWIN  10
#define NNEGS 5
#define NROW  60          // 10 pos + 50 neg rows per b
#define NTILE 4           // 64 padded rows / 16

__device__ __forceinline__ float softplus_f(float x) {
    // log(1+exp(x)) = max(x,0) + log1p(exp(-|x|))  (overflow safe)
    return fmaxf(x, 0.0f) + log1pf(expf(-fabsf(x)));
}

__global__ __launch_bounds__(256)
void pv_wmma_kernel(const int*   __restrict__ review_ids,   // [B]
                    const float* __restrict__ rwe,          // [B, W, 128]
                    const float* __restrict__ wmask,        // [B, W]
                    const int*   __restrict__ neg_idxs,     // [B, W*NNEG]
                    const float* __restrict__ wtab,         // [VOCAB, 128]
                    const float* __restrict__ rtab,         // [REVIEWS, 128]
                    float*       __restrict__ out_emb,      // [B, 128]
                    float*       __restrict__ out_loss,     // [B]
                    int Btot)
{
    const int lane   = threadIdx.x & 31;
    const int laneLo = lane & 15;       // M/N index within tile
    const int laneHi = lane >> 4;       // 0: K0/K1 half, 1: K2/K3 half
    const int wid    = threadIdx.x >> 5;

    int b = blockIdx.x * 8 + wid;
    if (b >= Btot) b = Btot - 1;        // duplicate work writes identical data

    __shared__ __align__(16) float lds_v[8][EMB];   // review emb per wave
    __shared__ __align__(16) float lds_s[8][64];    // 64 scores per wave

    // ---- gather review embedding, emit it, stash for B-fragment builds ----
    const int ridx = review_ids[b];
    float4 vfrag = ((const float4*)(rtab + (size_t)ridx * EMB))[lane];
    ((float4*)(out_emb + (size_t)b * EMB))[lane] = vfrag;
    ((float4*)lds_v[wid])[lane] = vfrag;
    __syncthreads();

    // ---- per-lane row pointers for the 4 A-tiles (rows 0..63) ----
    const float* rowp[NTILE];
    {
        // tile 0 row: laneLo in [0,16): pos rows 0..9, neg rows 10..15
        int r0 = laneLo;
        rowp[0] = (r0 < WWIN)
            ? rwe + ((size_t)b * WWIN + r0) * EMB
            : wtab + (size_t)neg_idxs[(size_t)b * (WWIN * NNEGS) + (r0 - WWIN)] * EMB;
        // tiles 1,2: pure neg rows
        rowp[1] = wtab + (size_t)neg_idxs[(size_t)b * (WWIN * NNEGS) + (16 + laneLo - WWIN)] * EMB;
        rowp[2] = wtab + (size_t)neg_idxs[(size_t)b * (WWIN * NNEGS) + (32 + laneLo - WWIN)] * EMB;
        // tile 3: rows 48..59 neg, 60..63 zero padding
        int r3 = 48 + laneLo;
        int n3 = (r3 < NROW) ? neg_idxs[(size_t)b * (WWIN * NNEGS) + (r3 - WWIN)] : 0;
        rowp[3] = wtab + (size_t)n3 * EMB;
    }
    const float m3 = (48 + laneLo < NROW) ? 1.0f : 0.0f;   // pad-row mask, tile 3 only

    // fold per-half-wave K offset (lanes 16-31 carry K+2, K+3)
    const int koff = 2 * laneHi;
    const float* vp = lds_v[wid] + koff;

    // ---- K loop: D = A(16x4 f32) x B(4x16 f32) + C, column 0 of B = review emb ----
    v8f c0 = {}, c1 = {}, c2 = {}, c3 = {};
    for (int k = 0; k < EMB; k += 4) {
        v2f bv = *(const v2f*)(vp + k);
        v2f bfrag;
        bfrag.x = (laneLo == 0) ? bv.x : 0.0f;   // B[k+koff  , 0]
        bfrag.y = (laneLo == 0) ? bv.y : 0.0f;   // B[k+koff+1, 0]

        v2f a0 = *(const v2f*)(rowp[0] + koff + k);
        v2f a1 = *(const v2f*)(rowp[1] + koff + k);
        v2f a2 = *(const v2f*)(rowp[2] + koff + k);
        v2f a3 = *(const v2f*)(rowp[3] + koff + k);
        a3.x *= m3; a3.y *= m3;

        c0 = __builtin_amdgcn_wmma_f32_16x16x4_f32(false, a0, false, bfrag, (short)0, c0, false, false);
        c1 = __builtin_amdgcn_wmma_f32_16x16x4_f32(false, a1, false, bfrag, (short)0, c1, false, false);
        c2 = __builtin_amdgcn_wmma_f32_16x16x4_f32(false, a2, false, bfrag, (short)0, c2, false, false);
        c3 = __builtin_amdgcn_wmma_f32_16x16x4_f32(false, a3, false, bfrag, (short)0, c3, false, false);
    }

    // ---- column 0 of C lives in lanes 0 (M=0..7) and 16 (M=8..15) ----
    if (laneLo == 0) {
        #pragma unroll
        for (int j = 0; j < 8; ++j) {
            lds_s[wid][ 0 + laneHi * 8 + j] = c0[j];
            lds_s[wid][16 + laneHi * 8 + j] = c1[j];
            lds_s[wid][32 + laneHi * 8 + j] = c2[j];
            lds_s[wid][48 + laneHi * 8 + j] = c3[j];
        }
    }
    __syncthreads();

    // ---- loss: softplus(-pos) + sum softplus(neg), masked mean over window ----
    float contrib = 0.0f, mcontrib = 0.0f;
    if (lane < WWIN) {
        float lw = softplus_f(-lds_s[wid][lane]);
        #pragma unroll
        for (int j = 0; j < NNEGS; ++j)
            lw += softplus_f(lds_s[wid][WWIN + lane * NNEGS + j]);
        float mw = wmask[(size_t)b * WWIN + lane];
        contrib  = lw * mw;
        mcontrib = mw;
    }
    #pragma unroll
    for (int off = 16; off > 0; off >>= 1) {
        contrib  += __shfl_xor(contrib,  off, 32);
        mcontrib += __shfl_xor(mcontrib, off, 32);
    }
    if (lane == 0)
        out_loss[b] = contrib / fmaxf(mcontrib, 1.0f);
}

extern "C" void kernel_launch(void* const* d_in, const int* in_sizes, int n_in,
                              void* d_out, int out_size, void* d_ws, size_t ws_size,
                              hipStream_t stream) {
    const int*   review_ids = (const int*)  d_in[0];
    const float* rwe        = (const float*)d_in[1];
    const float* wmask      = (const float*)d_in[2];
    const int*   neg_idxs   = (const int*)  d_in[3];
    const float* wtab       = (const float*)d_in[4];
    const float* rtab       = (const float*)d_in[5];
    // d_in[6] = n_negs scalar (compile-time NNEGS here)

    const int B = in_sizes[0];
    float* out_emb  = (float*)d_out;
    float* out_loss = (float*)d_out + (size_t)B * EMB;

    const int blocks = (B + 7) / 8;     // 8 waves (8 b's) per 256-thread block
    hipLaunchKernelGGL(pv_wmma_kernel, dim3(blocks), dim3(256), 0, stream,
                       review_ids, rwe, wmask, neg_idxs, wtab, rtab,
                       out_emb, out_loss, B);
}